// Pseudo3DConv_62311385530411
// MI455X (gfx1250) — compile-verified
//
#include <hip/hip_runtime.h>
#include <math.h>

// ---------------------------------------------------------------------------
// Pseudo3DConv forward for MI455X (gfx1250, wave32).
// All GEMMs run on V_WMMA_F32_16X16X4_F32 (fp32 in / fp32 acc => exact dtype
// match with the fp32 reference). Gathers, biases, leaky-relu, softmax scale
// and the NP=8 neighbor mean are fused around the WMMA tiles.
// ---------------------------------------------------------------------------

#define NPTS 500
#define NP   8
#define NCOL (NPTS * NP)   // 4000
#define SLOPE 0.01f

typedef __attribute__((ext_vector_type(2))) float v2f;
typedef __attribute__((ext_vector_type(8))) float v8f;

// ---------------------------------------------------------------------------
// 1) KNN both directions + per-neighbor Euclidean distances.
//    One thread per query point; insertion-sorted top-8 (matches top_k tie
//    order: equal keys keep the earlier index first).
// ---------------------------------------------------------------------------
__global__ __launch_bounds__(128) void knn_kernel(
    const float* __restrict__ c,   // (500,3)
    const float* __restrict__ ct,  // (500,3)
    int*   __restrict__ inds,    float* __restrict__ norms,     // dir 0: q=c, r=ct
    int*   __restrict__ inds_tc, float* __restrict__ norms_tc)  // dir 1: q=ct, r=c
{
    int q   = blockIdx.x * blockDim.x + threadIdx.x;
    int dir = blockIdx.y;
    if (q >= NPTS) return;
    const float* Q = (dir == 0) ? c  : ct;
    const float* R = (dir == 0) ? ct : c;
    int*   oi = (dir == 0) ? inds  : inds_tc;
    float* on = (dir == 0) ? norms : norms_tc;

    float qx = Q[q * 3 + 0], qy = Q[q * 3 + 1], qz = Q[q * 3 + 2];
    float bd[NP]; int bi[NP];
#pragma unroll
    for (int k = 0; k < NP; ++k) { bd[k] = 3.4e38f; bi[k] = 0; }

    for (int r = 0; r < NPTS; ++r) {
        float dx = qx - R[r * 3 + 0];
        float dy = qy - R[r * 3 + 1];
        float dz = qz - R[r * 3 + 2];
        float d2 = dx * dx + dy * dy + dz * dz;
        if (d2 < bd[NP - 1]) {
            int k = NP - 1;
            while (k > 0 && d2 < bd[k - 1]) { bd[k] = bd[k - 1]; bi[k] = bi[k - 1]; --k; }
            bd[k] = d2; bi[k] = r;
        }
    }
#pragma unroll
    for (int k = 0; k < NP; ++k) {
        oi[q * NP + k] = bi[k];
        on[q * NP + k] = sqrtf(bd[k]);
    }
}

// ---------------------------------------------------------------------------
// 2) Global softmax over -norm for each of the two 4000-vectors.
//    One block per vector (blockIdx.x in {0,1}).
// ---------------------------------------------------------------------------
__global__ __launch_bounds__(256) void softmax2_kernel(
    const float* __restrict__ n0, float* __restrict__ w0,
    const float* __restrict__ n1, float* __restrict__ w1)
{
    const float* x = (blockIdx.x == 0) ? n0 : n1;
    float*       y = (blockIdx.x == 0) ? w0 : w1;
    __shared__ float red[256];
    int t = threadIdx.x;

    float m = -3.4e38f;
    for (int i = t; i < NCOL; i += 256) m = fmaxf(m, -x[i]);
    red[t] = m; __syncthreads();
    for (int s = 128; s > 0; s >>= 1) { if (t < s) red[t] = fmaxf(red[t], red[t + s]); __syncthreads(); }
    m = red[0]; __syncthreads();

    float sum = 0.f;
    for (int i = t; i < NCOL; i += 256) sum += expf(-x[i] - m);
    red[t] = sum; __syncthreads();
    for (int s = 128; s > 0; s >>= 1) { if (t < s) red[t] += red[t + s]; __syncthreads(); }
    float inv = 1.0f / red[0];

    for (int i = t; i < NCOL; i += 256) y[i] = expf(-x[i] - m) * inv;
}

// ---------------------------------------------------------------------------
// 3) cloud_feat = pconv2(lrelu(pconv1(c.T))) : (3x500) -> (64) -> (128x500).
//    Tiny, scalar-FMA shaped (K=3 then K=64); one block per point.
// ---------------------------------------------------------------------------
__global__ __launch_bounds__(128) void pconv_kernel(
    const float* __restrict__ c,                            // (500,3)
    const float* __restrict__ w1, const float* __restrict__ b1,  // 64x3, 64
    const float* __restrict__ w2, const float* __restrict__ b2,  // 128x64, 128
    float* __restrict__ out)                                // 128x500
{
    int j = blockIdx.x;
    __shared__ float h[64];
    float x0 = c[j * 3 + 0], x1 = c[j * 3 + 1], x2 = c[j * 3 + 2];
    int t = threadIdx.x;
    if (t < 64) {
        float v = w1[t * 3 + 0] * x0 + w1[t * 3 + 1] * x1 + w1[t * 3 + 2] * x2 + b1[t];
        h[t] = (v >= 0.f) ? v : SLOPE * v;
    }
    __syncthreads();
    float acc = b2[t];
#pragma unroll 8
    for (int k = 0; k < 64; ++k) acc = fmaf(w2[t * 64 + k], h[k], acc);
    out[t * NPTS + j] = acc;
}

// ---------------------------------------------------------------------------
// WMMA helper: one K=4 step of a 16x16 fp32 tile.
// ---------------------------------------------------------------------------
__device__ __forceinline__ v8f wmma4(v2f a, v2f b, v8f c) {
    return __builtin_amdgcn_wmma_f32_16x16x4_f32(false, a, false, b, (short)0, c, false, false);
}

// ---------------------------------------------------------------------------
// 4) Fused branch:  out(C3 x 500) = mean_8( (W2 @ lrelu(W1 @ src[:,idx] + b1)
//                                            + b2) * wsm )
//    One block = 16 columns (= 2 output points). 8 waves; stage1 builds the
//    C2x16 intermediate in LDS with WMMA, stage2 does the C3x16 output tiles
//    with WMMA, then a lane-group-of-8 shuffle butterfly does the NP mean.
// ---------------------------------------------------------------------------
template <int C1, int C2, int C3>
__global__ __launch_bounds__(256) void branch_kernel(
    const float* __restrict__ src,                 // C1 x 500
    const int*   __restrict__ idx,                 // 4000 gather indices
    const float* __restrict__ wsm,                 // 4000 softmax weights
    const float* __restrict__ W1, const float* __restrict__ B1,  // C2 x C1, C2
    const float* __restrict__ W2, const float* __restrict__ B2,  // C3 x C2, C3
    float* __restrict__ out)                       // C3 x 500
{
    __shared__ float T[C2 * 16];
    const int n0   = blockIdx.x * 16;
    const int lane = threadIdx.x & 31;
    const int wave = threadIdx.x >> 5;      // 0..7
    const int half = lane >> 4;
    const int lrow = lane & 15;
    const int gcol = n0 + lrow;
    const int cidx = idx[gcol];              // gather column for this lane's N

    // ---- stage 1: T = lrelu(W1 @ src[:,idx] + b1), tile rows 16 at a time
    for (int tile = wave; tile < C2 / 16; tile += 8) {
        const int m0 = tile * 16;
        v8f acc = {};
        for (int k0 = 0; k0 < C1; k0 += 4) {
            const int kb = k0 + 2 * half;
            v2f a, b;
            const float* ar = W1 + (m0 + lrow) * C1 + kb;
            a.x = ar[0];  a.y = ar[1];
            b.x = src[kb * NPTS + cidx];
            b.y = src[(kb + 1) * NPTS + cidx];
            acc = wmma4(a, b, acc);
        }
#pragma unroll
        for (int i = 0; i < 8; ++i) {
            const int row = m0 + i + 8 * half;
            float v = acc[i] + B1[row];
            v = (v >= 0.f) ? v : SLOPE * v;
            T[row * 16 + lrow] = v;
        }
    }
    __syncthreads();

    // ---- stage 2: (W2 @ T + b2) * wsm, then mean over 8 neighbor columns
    const float wcol = wsm[gcol];
    for (int tile = wave; tile < C3 / 16; tile += 8) {
        const int m0 = tile * 16;
        v8f acc = {};
        for (int k0 = 0; k0 < C2; k0 += 4) {
            const int kb = k0 + 2 * half;
            v2f a, b;
            const float* ar = W2 + (m0 + lrow) * C2 + kb;
            a.x = ar[0];  a.y = ar[1];
            b.x = T[kb * 16 + lrow];
            b.y = T[(kb + 1) * 16 + lrow];
            acc = wmma4(a, b, acc);
        }
#pragma unroll
        for (int i = 0; i < 8; ++i) {
            const int row = m0 + i + 8 * half;
            float v = (acc[i] + B2[row]) * wcol;
            v += __shfl_xor(v, 1, 32);
            v += __shfl_xor(v, 2, 32);
            v += __shfl_xor(v, 4, 32);
            if ((lrow & 7) == 0) {
                const int p = (n0 >> 3) + (lrow >> 3);   // output point index
                out[row * NPTS + p] = v * 0.125f;
            }
        }
    }
}

// ---------------------------------------------------------------------------
// 5) Fused head:
//    f2 = w_fc2 @ [sfp; imf] + b_fc2          (64 x 500, K=160)
//    f1 = w_fc1 @ [sf; cloud_feat] + b_fc1    (64 x 500, K=256)
//    out = w_fc @ lrelu([f2; f1]) + b_fc      (64 x 500, K=128)
//    One block = 16 columns. Waves 0-3 build f2, waves 4-7 build f1 (both
//    lrelu'd into LDS), then waves 0-3 do the final conv and store d_out.
// ---------------------------------------------------------------------------
__global__ __launch_bounds__(256) void head_kernel(
    const float* __restrict__ sf,   const float* __restrict__ sfp,
    const float* __restrict__ cfeat, const float* __restrict__ imf,
    const float* __restrict__ Wfc1, const float* __restrict__ bfc1,
    const float* __restrict__ Wfc2, const float* __restrict__ bfc2,
    const float* __restrict__ Wfc,  const float* __restrict__ bfc,
    float* __restrict__ out)
{
    __shared__ float L[128 * 16];
    const int n0   = blockIdx.x * 16;
    const int lane = threadIdx.x & 31;
    const int wave = threadIdx.x >> 5;
    const int half = lane >> 4;
    const int lrow = lane & 15;
    const int gcol = n0 + lrow;
    const int col  = (gcol < NPTS) ? gcol : (NPTS - 1);   // clamp edge tile

    if (wave < 4) {           // f2 tile rows [m0, m0+16), K = 160
        const int m0 = wave * 16;
        v8f acc = {};
        for (int k0 = 0; k0 < 160; k0 += 4) {
            const int kb = k0 + 2 * half;   // even, so kb/kb+1 never straddle 128
            v2f a, b;
            const float* ar = Wfc2 + (m0 + lrow) * 160 + kb;
            a.x = ar[0];  a.y = ar[1];
            b.x = (kb < 128) ? sfp[kb * NPTS + col] : imf[(kb - 128) * NPTS + col];
            b.y = (kb + 1 < 128) ? sfp[(kb + 1) * NPTS + col] : imf[(kb + 1 - 128) * NPTS + col];
            acc = wmma4(a, b, acc);
        }
#pragma unroll
        for (int i = 0; i < 8; ++i) {
            const int row = m0 + i + 8 * half;
            float v = acc[i] + bfc2[row];
            v = (v >= 0.f) ? v : SLOPE * v;
            L[row * 16 + lrow] = v;               // concat rows 0..63 = f2
        }
    } else {                  // f1 tile rows [m0, m0+16), K = 256
        const int m0 = (wave - 4) * 16;
        v8f acc = {};
        for (int k0 = 0; k0 < 256; k0 += 4) {
            const int kb = k0 + 2 * half;
            v2f a, b;
            const float* ar = Wfc1 + (m0 + lrow) * 256 + kb;
            a.x = ar[0];  a.y = ar[1];
            b.x = (kb < 128) ? sf[kb * NPTS + col] : cfeat[(kb - 128) * NPTS + col];
            b.y = (kb + 1 < 128) ? sf[(kb + 1) * NPTS + col] : cfeat[(kb + 1 - 128) * NPTS + col];
            acc = wmma4(a, b, acc);
        }
#pragma unroll
        for (int i = 0; i < 8; ++i) {
            const int row = m0 + i + 8 * half;
            float v = acc[i] + bfc1[row];
            v = (v >= 0.f) ? v : SLOPE * v;
            L[(64 + row) * 16 + lrow] = v;        // concat rows 64..127 = f1
        }
    }
    __syncthreads();

    if (wave < 4) {           // out = w_fc @ L + b_fc, K = 128
        const int m0 = wave * 16;
        v8f acc = {};
        for (int k0 = 0; k0 < 128; k0 += 4) {
            const int kb = k0 + 2 * half;
            v2f a, b;
            const float* ar = Wfc + (m0 + lrow) * 128 + kb;
            a.x = ar[0];  a.y = ar[1];
            b.x = L[kb * 16 + lrow];
            b.y = L[(kb + 1) * 16 + lrow];
            acc = wmma4(a, b, acc);
        }
#pragma unroll
        for (int i = 0; i < 8; ++i) {
            const int row = m0 + i + 8 * half;
            if (gcol < NPTS) out[row * NPTS + gcol] = acc[i] + bfc[row];
        }
    }
}

// ---------------------------------------------------------------------------
// Host-side launcher
// ---------------------------------------------------------------------------
extern "C" void kernel_launch(void* const* d_in, const int* in_sizes, int n_in,
                              void* d_out, int out_size, void* d_ws, size_t ws_size,
                              hipStream_t stream)
{
    (void)in_sizes; (void)n_in; (void)out_size; (void)ws_size;

    const float* imf       = (const float*)d_in[0];   // (32,500)
    const float* cloud     = (const float*)d_in[1];   // (500,3)
    const float* cloud_tar = (const float*)d_in[2];   // (500,3)
    const float* w_conv1   = (const float*)d_in[3];   // 64x32
    const float* b_conv1   = (const float*)d_in[4];
    const float* w_conv2   = (const float*)d_in[5];   // 128x64
    const float* b_conv2   = (const float*)d_in[6];
    const float* w_psconv1 = (const float*)d_in[7];   // 256x128
    const float* b_psconv1 = (const float*)d_in[8];
    const float* w_psconv2 = (const float*)d_in[9];   // 128x256
    const float* b_psconv2 = (const float*)d_in[10];
    const float* w_pconv1  = (const float*)d_in[11];  // 64x3
    const float* b_pconv1  = (const float*)d_in[12];
    const float* w_pconv2  = (const float*)d_in[13];  // 128x64
    const float* b_pconv2  = (const float*)d_in[14];
    const float* w_fc1     = (const float*)d_in[15];  // 64x256
    const float* b_fc1     = (const float*)d_in[16];
    const float* w_fc2     = (const float*)d_in[17];  // 64x160
    const float* b_fc2     = (const float*)d_in[18];
    const float* w_fc      = (const float*)d_in[19];  // 64x128
    const float* b_fc      = (const float*)d_in[20];

    char* ws = (char*)d_ws;
    int*   inds     = (int*)  (ws + 0);
    int*   inds_tc  = (int*)  (ws + 16000);
    float* norms    = (float*)(ws + 32000);
    float* norms_tc = (float*)(ws + 48000);
    float* wsm      = (float*)(ws + 64000);
    float* wsm_tc   = (float*)(ws + 80000);
    float* cfeat    = (float*)(ws + 96000);            // 128x500
    float* sfbuf    = (float*)(ws + 96000 + 256000);   // 128x500
    float* sfpbuf   = (float*)(ws + 96000 + 512000);   // 128x500

    knn_kernel<<<dim3(4, 2), 128, 0, stream>>>(cloud, cloud_tar,
                                               inds, norms, inds_tc, norms_tc);
    softmax2_kernel<<<2, 256, 0, stream>>>(norms, wsm, norms_tc, wsm_tc);
    pconv_kernel<<<NPTS, 128, 0, stream>>>(cloud, w_pconv1, b_pconv1,
                                           w_pconv2, b_pconv2, cfeat);
    branch_kernel<32, 64, 128><<<NCOL / 16, 256, 0, stream>>>(
        imf, inds, wsm, w_conv1, b_conv1, w_conv2, b_conv2, sfbuf);
    branch_kernel<128, 256, 128><<<NCOL / 16, 256, 0, stream>>>(
        cfeat, inds_tc, wsm_tc, w_psconv1, b_psconv1, w_psconv2, b_psconv2, sfpbuf);
    head_kernel<<<(NPTS + 15) / 16, 256, 0, stream>>>(
        sfbuf, sfpbuf, cfeat, imf,
        w_fc1, b_fc1, w_fc2, b_fc2, w_fc, b_fc, (float*)d_out);
}